// RandomNetwork_64063732187512
// MI455X (gfx1250) — compile-verified
//
#include <hip/hip_runtime.h>
#include <stdint.h>
#include <math.h>

#define NNODES 32
#define CCH 64
#define BATCH 8
#define HO 56
#define HW2 (HO * HO)                       // 3136
#define NODE_ELEMS (BATCH * CCH * HW2)      // 1,605,632 floats per node output
#define IN_H 112
#define IN_ELEMS (BATCH * CCH * IN_H * IN_H)
#define NCOLS (BATCH * HW2)                 // 25088 GEMM columns
#define NTILES (NCOLS / 16)                 // 1568 column tiles (exact)

typedef __attribute__((ext_vector_type(2))) float v2f;
typedef __attribute__((ext_vector_type(8))) float v8f;

struct PredList { int n; int idx[32]; };

// ---------------- Kernel 0: fold BN params into per-channel scale/bias ----------------
// scale = gamma / sqrt(var + eps); bias = beta - mean * scale   (for all 32 nodes x 64 ch)
__global__ void bn_fold_kernel(const float* __restrict__ gamma, const float* __restrict__ beta,
                               const float* __restrict__ mean,  const float* __restrict__ var,
                               float* __restrict__ scale, float* __restrict__ bias) {
  int i = blockIdx.x * blockDim.x + threadIdx.x;
  if (i >= NNODES * CCH) return;
  float s = gamma[i] / sqrtf(var[i] + 1e-5f);
  scale[i] = s;
  bias[i]  = beta[i] - mean[i] * s;
}

// ---------------- Kernel 1: aggregate (sigmoid-weighted) + ReLU ----------------
__global__ void agg_relu_kernel(const float* __restrict__ x,
                                const float* __restrict__ outs,     // base of all node outputs
                                const float* __restrict__ agg_row,  // agg_w + node*32
                                PredList pl, float* __restrict__ t, int total) {
  int i = blockIdx.x * blockDim.x + threadIdx.x;
  if (i >= total) return;
  float v;
  if (pl.n == 0) {
    v = x[i];
  } else if (pl.n == 1) {
    v = outs[(size_t)pl.idx[0] * NODE_ELEMS + i];
  } else {
    float acc = 0.f;
    for (int p = 0; p < pl.n; ++p) {
      float w = 1.f / (1.f + expf(-agg_row[p]));
      acc += w * outs[(size_t)pl.idx[p] * NODE_ELEMS + i];
    }
    v = acc;
  }
  t[i] = v > 0.f ? v : 0.f;
}

// ---------------- Kernel 2: depthwise 3x3, padding 1, stride s ----------------
__global__ void dw_kernel(const float* __restrict__ t, const float* __restrict__ w9,
                          float* __restrict__ d, int stride, int Hin) {
  int i = blockIdx.x * blockDim.x + threadIdx.x;
  if (i >= NODE_ELEMS) return;
  int wo = i % HO;
  int ho = (i / HO) % HO;
  int c  = (i / HW2) % CCH;
  int n  = i / (HW2 * CCH);
  const float* tin = t + ((size_t)(n * CCH + c)) * Hin * Hin;
  const float* w = w9 + c * 9;
  int hi0 = ho * stride - 1, wi0 = wo * stride - 1;
  float s = 0.f;
#pragma unroll
  for (int kh = 0; kh < 3; ++kh) {
    int hi = hi0 + kh;
    if (hi < 0 || hi >= Hin) continue;
#pragma unroll
    for (int kw = 0; kw < 3; ++kw) {
      int wi = wi0 + kw;
      if (wi < 0 || wi >= Hin) continue;
      s += w[kh * 3 + kw] * tin[hi * Hin + wi];
    }
  }
  d[i] = s;
}

// ---------------- Kernel 3: pointwise 64x64 GEMM (fp32 WMMA) + folded BN ----------------
__global__ __launch_bounds__(256)
void pw_bn_wmma_kernel(const float* __restrict__ d,     // [8,64,3136]
                       const float* __restrict__ pw,    // [64,64] row-major (o,c)
                       const float* __restrict__ scale, // [64] folded BN scale
                       const float* __restrict__ bias,  // [64] folded BN bias
                       float* __restrict__ out) {
  int wave = (blockIdx.x * blockDim.x + threadIdx.x) >> 5;
  int lane = threadIdx.x & 31;
  if (wave >= NTILES) return;
  int col0 = wave << 4;
  int n  = col0 / HW2;
  int hw = col0 - n * HW2 + (lane & 15);
  const float* dn = d + (size_t)n * (CCH * HW2);
  int koff = (lane >> 4) << 1;  // lanes 0-15: K rows {0,1}; lanes 16-31: K rows {2,3}

  // Preload 16 B-fragments (each a 4x16 fp32 chunk of the activation matrix)
  v2f b[16];
#pragma unroll
  for (int k = 0; k < 16; ++k) {
    const float* p = dn + (size_t)(4 * k + koff) * HW2 + hw;
    b[k].x = p[0];
    b[k].y = p[HW2];
  }

  int arow = lane & 15;
#pragma unroll
  for (int m = 0; m < 4; ++m) {
    v8f acc = {};
    const float* ap = pw + (16 * m + arow) * CCH + koff;
#pragma unroll
    for (int k = 0; k < 16; ++k) {
      v2f a;
      a.x = ap[4 * k];
      a.y = ap[4 * k + 1];
      acc = __builtin_amdgcn_wmma_f32_16x16x4_f32(
          /*neg_a=*/false, a, /*neg_b=*/false, b[k],
          /*c_mod=*/(short)0, acc, /*reuse_a=*/false, /*reuse_b=*/false);
    }
    // C/D layout: VGPR r -> row (16*m + r) for lanes 0-15, (16*m + r + 8) for lanes 16-31
    int obase = 16 * m + ((lane >> 4) << 3);
    float* op = out + (size_t)n * (CCH * HW2) + hw;
#pragma unroll
    for (int r = 0; r < 8; ++r) {
      int o = obase + r;
      op[(size_t)o * HW2] = acc[r] * scale[o] + bias[o];
    }
  }
}

// ---------------- Kernel 4: mean over FINAL nodes ----------------
__global__ void mean_kernel(const float* __restrict__ outs, PredList fl,
                            float* __restrict__ out) {
  int i = blockIdx.x * blockDim.x + threadIdx.x;
  if (i >= NODE_ELEMS) return;
  float s = 0.f;
  for (int f = 0; f < fl.n; ++f) s += outs[(size_t)fl.idx[f] * NODE_ELEMS + i];
  out[i] = s * (1.f / fl.n);
}

// ---------------- Host: reproduce numpy default_rng(0).random((32,32)) < 0.25 ----------------
static void build_dag(bool adj[NNODES][NNODES]) {
  // SeedSequence(0): entropy words = [0], pool_size = 4
  uint32_t pool[4];
  uint32_t hc = 0x43b0d7e5u;  // INIT_A
  auto hashmix = [&hc](uint32_t value) -> uint32_t {
    value ^= hc;
    hc *= 0x931e8875u;        // MULT_A
    value *= hc;
    value ^= value >> 16;
    return value;
  };
  auto mix = [](uint32_t x, uint32_t y) -> uint32_t {
    uint32_t r = (x * 0xca01f9ddu) ^ (y * 0x4973f715u);  // MIX_MULT_L/R
    r ^= r >> 16;
    return r;
  };
  for (int i = 0; i < 4; ++i) pool[i] = hashmix(0u);  // entropy[0]=0, rest pad 0
  for (int s = 0; s < 4; ++s)
    for (int d = 0; d < 4; ++d)
      if (s != d) pool[d] = mix(pool[d], hashmix(pool[s]));

  // generate_state(4, uint64) -> 8 uint32 words (little-endian pairing)
  uint32_t hc2 = 0x8b51f9ddu;  // INIT_B
  uint32_t st[8];
  for (int i = 0; i < 8; ++i) {
    uint32_t dv = pool[i & 3];
    dv ^= hc2;
    hc2 *= 0x58f38dedu;        // MULT_B
    dv *= hc2;
    dv ^= dv >> 16;
    st[i] = dv;
  }
  uint64_t w[4];
  for (int i = 0; i < 4; ++i)
    w[i] = (uint64_t)st[2 * i] | ((uint64_t)st[2 * i + 1] << 32);

  // PCG64 (XSL-RR 128/64, setseq): seed = (w0<<64)|w1, inc-seed = (w2<<64)|w3
  typedef unsigned __int128 u128;
  const u128 MULT = ((u128)0x2360ed051fc65da4ULL << 64) | 0x4385df649fccf645ULL;
  u128 initstate = ((u128)w[0] << 64) | w[1];
  u128 initseq   = ((u128)w[2] << 64) | w[3];
  u128 state = 0;
  u128 inc = (initseq << 1) | 1;
  state = state * MULT + inc;
  state += initstate;
  state = state * MULT + inc;

  for (int i = 0; i < NNODES; ++i)
    for (int j = 0; j < NNODES; ++j) {
      state = state * MULT + inc;  // step, then output (128-bit variant)
      uint64_t hi = (uint64_t)(state >> 64), lo = (uint64_t)state;
      unsigned rot = (unsigned)(hi >> 58);
      uint64_t xv = hi ^ lo;
      uint64_t o = rot ? ((xv >> rot) | (xv << (64 - rot))) : xv;
      double dv = (double)(o >> 11) * (1.0 / 9007199254740992.0);
      adj[i][j] = dv < 0.25;
    }
}

extern "C" void kernel_launch(void* const* d_in, const int* in_sizes, int n_in,
                              void* d_out, int out_size, void* d_ws, size_t ws_size,
                              hipStream_t stream) {
  const float* x     = (const float*)d_in[0];
  const float* dw    = (const float*)d_in[1];
  const float* pw    = (const float*)d_in[2];
  const float* gamma = (const float*)d_in[3];
  const float* beta  = (const float*)d_in[4];
  const float* mean  = (const float*)d_in[5];
  const float* var   = (const float*)d_in[6];
  const float* agg_w = (const float*)d_in[7];

  // Workspace layout (floats):
  //   32 node outputs | tmp_y (112^2 sized) | tmp_d (56^2 sized) | bn_scale[2048] | bn_bias[2048]
  float* outs     = (float*)d_ws;
  float* tmp_y    = outs + (size_t)NNODES * NODE_ELEMS;
  float* tmp_d    = tmp_y + IN_ELEMS;
  float* bn_scale = tmp_d + NODE_ELEMS;
  float* bn_bias  = bn_scale + NNODES * CCH;

  // Reconstruct the static DAG (matches numpy default_rng(0))
  bool adj[NNODES][NNODES];
  build_dag(adj);
  PredList preds[NNODES];
  int ispan[NNODES];
  for (int j = 0; j < NNODES; ++j) {
    preds[j].n = 0;
    for (int i = 0; i < j; ++i)
      if (adj[i][j]) preds[j].idx[preds[j].n++] = i;
  }
  for (int i = 0; i < NNODES; ++i) {
    ispan[i] = NNODES;
    int mx = -1;
    for (int j = i + 1; j < NNODES; ++j)
      if (adj[i][j]) mx = j;
    if (mx >= 0) ispan[i] = mx;
  }
  PredList finals;
  finals.n = 0;
  for (int i = 0; i < NNODES; ++i)
    if (ispan[i] >= NNODES - 1) finals.idx[finals.n++] = i;

  const int TB = 256;

  // Fold BN params once for all nodes (2048 channels total)
  bn_fold_kernel<<<(NNODES * CCH + TB - 1) / TB, TB, 0, stream>>>(
      gamma, beta, mean, var, bn_scale, bn_bias);

  for (int i = 0; i < NNODES; ++i) {
    int total_in = (preds[i].n == 0) ? IN_ELEMS : NODE_ELEMS;
    agg_relu_kernel<<<(total_in + TB - 1) / TB, TB, 0, stream>>>(
        x, outs, agg_w + (size_t)i * NNODES, preds[i], tmp_y, total_in);

    int stride = (preds[i].n == 0) ? 2 : 1;
    int Hin    = (preds[i].n == 0) ? IN_H : HO;
    dw_kernel<<<(NODE_ELEMS + TB - 1) / TB, TB, 0, stream>>>(
        tmp_y, dw + (size_t)i * CCH * 9, tmp_d, stride, Hin);

    pw_bn_wmma_kernel<<<(NTILES * 32 + TB - 1) / TB, TB, 0, stream>>>(
        tmp_d, pw + (size_t)i * CCH * CCH,
        bn_scale + (size_t)i * CCH, bn_bias + (size_t)i * CCH,
        outs + (size_t)i * NODE_ELEMS);
  }

  mean_kernel<<<(NODE_ELEMS + TB - 1) / TB, TB, 0, stream>>>(outs, finals, (float*)d_out);
}